// DacResidualVectorQuantize_44968307589253
// MI455X (gfx1250) — compile-verified
//
#include <hip/hip_runtime.h>

typedef __attribute__((ext_vector_type(2))) float v2f;
typedef __attribute__((ext_vector_type(8))) float v8f;

static constexpr int kB   = 16;    // batch
static constexpr int kD   = 1024;  // latent dim
static constexpr int kT   = 2048;  // time steps
static constexpr int kN   = 9;     // codebooks
static constexpr int kCD  = 8;     // codebook dim
static constexpr int kCS  = 1024;  // codebook size
static constexpr int kTILE = 16;   // time columns per wave
static constexpr int kWAVES = 4;   // waves per workgroup (one per SIMD32)
static constexpr int kCOLS = kTILE * kWAVES;  // 64 columns per WG
static constexpr int kRS   = 76;   // LDS row stride (floats): 16B-aligned rows,
                                   // bank shift 24 for row-delta 2, 32 for delta 8

// ---------------- prep kernel 1: weight-norm of in_v rows (norm over D) ----------
__global__ void prep_win(const float* __restrict__ in_v, const float* __restrict__ in_g,
                         float* __restrict__ winN) {
  __shared__ float red[256];
  int row = blockIdx.x;                    // 0..N*CD-1
  const float* v = in_v + (size_t)row * kD;
  float s = 0.f;
  for (int d = threadIdx.x; d < kD; d += 256) { float x = v[d]; s += x * x; }
  red[threadIdx.x] = s;
  __syncthreads();
  for (int k = 128; k > 0; k >>= 1) {
    if (threadIdx.x < k) red[threadIdx.x] += red[threadIdx.x + k];
    __syncthreads();
  }
  float sc = in_g[row] * rsqrtf(red[0]);
  float* o = winN + (size_t)row * kD;
  for (int d = threadIdx.x; d < kD; d += 256) o[d] = v[d] * sc;
}

// ---------------- prep kernel 2: weight-norm of out_v rows + l2n codebooks (len 8) ----
__global__ void prep_small(const float* __restrict__ out_v, const float* __restrict__ out_g,
                           const float* __restrict__ cbs,
                           float* __restrict__ woutN, float* __restrict__ cbN,
                           float* __restrict__ loss_out) {
  int r = blockIdx.x * blockDim.x + threadIdx.x;
  if (r == 0) { loss_out[0] = 0.f; loss_out[1] = 0.f; }
  int nw = kN * kD;
  int nc = kN * kCS;
  if (r < nw) {
    const float* v = out_v + (size_t)r * kCD;
    float s = 0.f;
#pragma unroll
    for (int k = 0; k < kCD; ++k) s += v[k] * v[k];
    float sc = out_g[r] * rsqrtf(s);
    float* o = woutN + (size_t)r * kCD;
#pragma unroll
    for (int k = 0; k < kCD; ++k) o[k] = v[k] * sc;
  } else if (r < nw + nc) {
    int rr = r - nw;
    const float* v = cbs + (size_t)rr * kCD;
    float s = 0.f;
#pragma unroll
    for (int k = 0; k < kCD; ++k) s += v[k] * v[k];
    float n = fmaxf(sqrtf(s), 1e-12f);
    float sc = 1.0f / n;
    float* o = cbN + (size_t)rr * kCD;
#pragma unroll
    for (int k = 0; k < kCD; ++k) o[k] = v[k] * sc;
  }
}

// z_e K-step: one WMMA f32 16x16x4 accumulating into ACC (independent per ACC)
#define ZE_STEP(ACC, KK)                                                        \
  {                                                                             \
    float2 wv = *(const float2*)(wi + (size_t)mr * kD + (KK) + koff);           \
    v2f a;                                                                      \
    a.x = amask ? wv.x : 0.f;                                                   \
    a.y = amask ? wv.y : 0.f;                                                   \
    v2f bb;                                                                     \
    bb.x = res[((KK) + koff) * kRS + wcol + m];                                 \
    bb.y = res[((KK) + koff + 1) * kRS + wcol + m];                             \
    ACC = __builtin_amdgcn_wmma_f32_16x16x4_f32(false, a, false, bb, (short)0,  \
                                                ACC, false, false);             \
  }

// ---------------- main fused RVQ kernel: 4 waves, wave = 16 cols, residual in LDS ----
__global__ __launch_bounds__(128) void rvq_main(
    const float* __restrict__ hidden, const float* __restrict__ in_b,
    const float* __restrict__ out_b, const float* __restrict__ cbs,
    const float* __restrict__ winN, const float* __restrict__ woutN,
    const float* __restrict__ cbN,
    float* __restrict__ quant_out, float* __restrict__ idx_out,
    float* __restrict__ lat_out, float* __restrict__ loss_out) {
  extern __shared__ float smem[];
  float* res   = smem;                               // [kD][kRS], cols 0..63 valid
  float* ze_b  = smem + kD * kRS;                    // [kWAVES][kCD][kTILE]
  float* zq_b  = ze_b + kWAVES * kCD * kTILE;        // [kWAVES][kCD][kTILE]
  int*   idx_b = (int*)(zq_b + kWAVES * kCD * kTILE);// [kWAVES][kTILE]

  const int lane = threadIdx.x & 31;        // wave32 lane
  const int wave = threadIdx.x >> 5;        // 0..3, one per SIMD
  const int wcol = wave * kTILE;            // this wave's column slice in LDS
  const int m    = lane & 15;               // fragment row/col index within 16
  const int koff = (lane >> 4) << 1;        // 0 or 2 (K sub-offset per ISA layout)
  const bool amask = (m < kCD);
  const int mr   = amask ? m : 0;

  const int wg = blockIdx.x;
  const int b  = wg / (kT / kCOLS);
  const int tw = (wg % (kT / kCOLS)) * kCOLS + wcol;  // global column base (wave)
  const float* hidB = hidden + (size_t)b * kD * kT;

  float* ze_w = ze_b + wave * kCD * kTILE;
  float* zq_w = zq_b + wave * kCD * kTILE;
  int*   idx_w = idx_b + wave * kTILE;

  // ---- phase 0: residual tile <- hidden tile via async global->LDS b128 ----
  for (int it = lane; it < kD * 4; it += 32) {
    int d = it >> 2, q = (it & 3) << 2;
    unsigned ldsb = (unsigned)(size_t)(const void*)(&res[d * kRS + wcol + q]);
    unsigned long long ga =
        (unsigned long long)(const void*)(hidB + (size_t)d * kT + tw + q);
    asm volatile("global_load_async_to_lds_b128 %0, %1, off"
                 :: "v"(ldsb), "v"(ga) : "memory");
  }
  asm volatile("s_wait_asynccnt 0" ::: "memory");

  float loss_acc = 0.f;

  for (int i = 0; i < kN; ++i) {
    const float* wi = winN  + (size_t)i * kCD * kD;   // [8][1024]
    const float* wo = woutN + (size_t)i * kD * kCD;   // [1024][8]
    const float* cn = cbN   + (size_t)i * kCS * kCD;  // [1024][8] normalized
    const float* cb = cbs   + (size_t)i * kCS * kCD;  // [1024][8] raw
    __builtin_prefetch(cn, 0, 1);
    __builtin_prefetch(wo, 0, 1);

    // ---- z_e = w_in(8x1024, padded to 16) * res(1024x16), 4 independent acc ----
    v8f acc0 = {0.f, 0.f, 0.f, 0.f, 0.f, 0.f, 0.f, 0.f};
    v8f acc1 = acc0, acc2 = acc0, acc3 = acc0;
    for (int k = 0; k < kD; k += 16) {
      ZE_STEP(acc0, k);
      ZE_STEP(acc1, k + 4);
      ZE_STEP(acc2, k + 8);
      ZE_STEP(acc3, k + 12);
    }
    v8f acc = (acc0 + acc1) + (acc2 + acc3);

    // bias + stash z_e (rows 0..7 live in lanes 0..15) + emit projected latents
    float ze[kCD];
#pragma unroll
    for (int v = 0; v < kCD; ++v) ze[v] = 0.f;
    if (lane < 16) {
#pragma unroll
      for (int v = 0; v < kCD; ++v) {
        float z = acc[v] + in_b[i * kCD + v];
        ze[v] = z;
        ze_w[v * kTILE + m] = z;
        lat_out[(size_t)b * (kN * kCD * kT) + (size_t)(i * kCD + v) * kT + tw + m] = z;
      }
    }

    // ---- similarity: sim(16 cols x 1024 codes) = z_e^T(16x8) * cn^T(8x1024) ----
    v2f sa0, sa1;
    sa0.x = ze_w[(koff + 0) * kTILE + m];
    sa0.y = ze_w[(koff + 1) * kTILE + m];
    sa1.x = ze_w[(4 + koff + 0) * kTILE + m];
    sa1.y = ze_w[(4 + koff + 1) * kTILE + m];
    float best[kCD];
    int   bidx[kCD];
#pragma unroll
    for (int v = 0; v < kCD; ++v) { best[v] = -3.4e38f; bidx[v] = 0; }
    for (int tile = 0; tile < kCS / 16; ++tile) {
      const float* cr = cn + (size_t)(tile * 16 + m) * kCD;
      float2 c0 = *(const float2*)(cr + koff);
      float2 c1 = *(const float2*)(cr + 4 + koff);
      v2f b0; b0.x = c0.x; b0.y = c0.y;
      v2f b1; b1.x = c1.x; b1.y = c1.y;
      v8f c = {0.f, 0.f, 0.f, 0.f, 0.f, 0.f, 0.f, 0.f};
      c = __builtin_amdgcn_wmma_f32_16x16x4_f32(false, sa0, false, b0, (short)0, c,
                                                false, false);
      c = __builtin_amdgcn_wmma_f32_16x16x4_f32(false, sa1, false, b1, (short)0, c,
                                                false, false);
      int code = tile * 16 + m;
#pragma unroll
      for (int v = 0; v < kCD; ++v) {
        float s = c[v];
        if (s > best[v]) { best[v] = s; bidx[v] = code; }
      }
    }
    // butterfly argmax within each 16-lane half (tie -> lowest index, matches argmin)
#pragma unroll
    for (int v = 0; v < kCD; ++v) {
      float bv = best[v];
      int ix = bidx[v];
#pragma unroll
      for (int off = 1; off < 16; off <<= 1) {
        float ov = __shfl_xor(bv, off);
        int   oi = __shfl_xor(ix, off);
        if (ov > bv || (ov == bv && oi < ix)) { bv = ov; ix = oi; }
      }
      if (lane == 0)  idx_w[v] = ix;       // columns 0..7
      if (lane == 16) idx_w[v + 8] = ix;   // columns 8..15
    }

    // ---- gather z_q, emit indices ----
    if (lane < 16) {
      int code = idx_w[m];
      idx_out[(size_t)b * (kN * kT) + (size_t)i * kT + tw + m] = (float)code;
      const float* crow = cb + (size_t)code * kCD;
#pragma unroll
      for (int o = 0; o < kCD; ++o) zq_w[o * kTILE + m] = crow[o];
    }

    // ---- loss: sum (z_e - z_q)^2  (commit == cb numerically) ----
    if (lane < 16) {
#pragma unroll
      for (int o = 0; o < kCD; ++o) {
        float dlt = ze[o] - zq_w[o * kTILE + m];
        loss_acc += dlt * dlt;
      }
    }

    // ---- out-projection: res -= w_out(1024x8) * z_q(8x16) + out_b ----
    v2f qb0, qb1;
    qb0.x = zq_w[(koff + 0) * kTILE + m];
    qb0.y = zq_w[(koff + 1) * kTILE + m];
    qb1.x = zq_w[(4 + koff + 0) * kTILE + m];
    qb1.y = zq_w[(4 + koff + 1) * kTILE + m];
    for (int tile = 0; tile < kD / 16; ++tile) {
      const float* wr = wo + (size_t)(tile * 16 + m) * kCD;
      float2 w0 = *(const float2*)(wr + koff);
      float2 w1 = *(const float2*)(wr + 4 + koff);
      v2f a0; a0.x = w0.x; a0.y = w0.y;
      v2f a1; a1.x = w1.x; a1.y = w1.y;
      v8f c = {0.f, 0.f, 0.f, 0.f, 0.f, 0.f, 0.f, 0.f};
      c = __builtin_amdgcn_wmma_f32_16x16x4_f32(false, a0, false, qb0, (short)0, c,
                                                false, false);
      c = __builtin_amdgcn_wmma_f32_16x16x4_f32(false, a1, false, qb1, (short)0, c,
                                                false, false);
      int dbase = tile * 16 + ((lane >> 4) << 3);
#pragma unroll
      for (int v = 0; v < kCD; ++v) {
        int d = dbase + v;
        float q = c[v] + out_b[i * kD + d];
        res[d * kRS + wcol + m] -= q;
      }
    }
  }

  // ---- loss reduction (per wave) -> global atomics ----
  for (int off = 16; off > 0; off >>= 1) loss_acc += __shfl_xor(loss_acc, off);
  if (lane == 0) {
    float v = loss_acc * (1.0f / ((float)kB * kCD * kT));
    atomicAdd(&loss_out[0], v);
    atomicAdd(&loss_out[1], v);
  }

  // ---- quant = hidden - residual_final ----
  for (int it = lane; it < kD * 4; it += 32) {
    int d = it >> 2, q = (it & 3) << 2;
    float4 h = *(const float4*)(hidB + (size_t)d * kT + tw + q);
    float4 r = *(const float4*)(&res[d * kRS + wcol + q]);
    float4 o;
    o.x = h.x - r.x; o.y = h.y - r.y; o.z = h.z - r.z; o.w = h.w - r.w;
    *(float4*)(quant_out + (size_t)b * kD * kT + (size_t)d * kT + tw + q) = o;
  }
}

extern "C" void kernel_launch(void* const* d_in, const int* in_sizes, int n_in,
                              void* d_out, int out_size, void* d_ws, size_t ws_size,
                              hipStream_t stream) {
  (void)in_sizes; (void)n_in; (void)out_size; (void)ws_size;
  const float* hidden = (const float*)d_in[0];
  const float* in_v   = (const float*)d_in[1];
  const float* in_g   = (const float*)d_in[2];
  const float* in_b   = (const float*)d_in[3];
  const float* out_v  = (const float*)d_in[4];
  const float* out_g  = (const float*)d_in[5];
  const float* out_b  = (const float*)d_in[6];
  const float* cbs    = (const float*)d_in[7];
  // d_in[8] = n_quantizers (fixed 9 in this problem)

  float* quant = (float*)d_out;
  float* idxo  = quant + (size_t)kB * kD * kT;          // [B,N,T]
  float* lato  = idxo + (size_t)kB * kN * kT;           // [B,N*CD,T]
  float* losso = lato + (size_t)kB * kN * kCD * kT;     // commit, cb

  float* winN  = (float*)d_ws;                          // [N,CD,D]
  float* woutN = winN + (size_t)kN * kCD * kD;          // [N,D,CD]
  float* cbN   = woutN + (size_t)kN * kD * kCD;         // [N,CS,CD]

  prep_win<<<kN * kCD, 256, 0, stream>>>(in_v, in_g, winN);
  int rows = kN * kD + kN * kCS;
  prep_small<<<(rows + 255) / 256, 256, 0, stream>>>(out_v, out_g, cbs, woutN, cbN, losso);

  size_t lds = (size_t)(kD * kRS + 2 * kWAVES * kCD * kTILE + kWAVES * kTILE)
               * sizeof(float);
  rvq_main<<<dim3(kB * (kT / kCOLS)), dim3(kWAVES * 32), lds, stream>>>(
      hidden, in_b, out_b, cbs, winN, woutN, cbN, quant, idxo, lato, losso);
}